// CTMBridge_70016556859903
// MI455X (gfx1250) — compile-verified
//
#include <hip/hip_runtime.h>
#include <hip/hip_bf16.h>
#include <math.h>

// ---------------------------------------------------------------------------
// CTMBridge for MI455X (gfx1250, wave32).
// Shapes: B=2, T=2048, D=1024, S=64, WD=256.  All fp32 (reference is fp32).
// GEMM stages use V_WMMA_F32_16X16X4_F32 (fp32 matrix pipe).  Large GEMMs use
// 4-wave workgroups that stage the shared 16-row A tile into LDS via
// GLOBAL_LOAD_ASYNC_TO_LDS_B128 (ASYNCcnt path), giving 4-64x A-operand reuse
// out of LDS instead of L2.  The T-scan is decomposed into chunk partial-sums
// (WMMA), a chunk-prefix pass, and an in-chunk register-resident sweep.
// ---------------------------------------------------------------------------

#define CB   2
#define CT   2048
#define CD   1024
#define CS   64
#define CWD  256
#define CM   (CB * CT)          // 4096 tokens
#define NC   16                 // chunks per batch
#define CH   (CT / NC)          // 128 tokens per chunk
#define KC   256                // K staging chunk for GEMMs
#define KPAD 4                  // LDS row pad (floats) -> conflict-free ds_load
#define FEPS 1.1920928955078125e-07f  // float32 eps (jnp.finfo(f32).eps)

typedef __attribute__((ext_vector_type(2))) float v2f;
typedef __attribute__((ext_vector_type(8))) float v8f;
typedef int v4i __attribute__((vector_size(16)));

static __device__ __forceinline__ v8f wmma_f32_16x16x4(v2f a, v2f b, v8f c) {
  // (neg_a, A, neg_b, B, c_mod, C, reuse_a, reuse_b)
  return __builtin_amdgcn_wmma_f32_16x16x4_f32(false, a, false, b, (short)0, c,
                                               false, false);
}

// --- async global -> LDS staging (gfx1250 ASYNCcnt path), with fallback ----
#if __has_builtin(__builtin_amdgcn_global_load_async_to_lds_b128) && \
    __has_builtin(__builtin_amdgcn_s_wait_asynccnt)
#define CTM_ASYNC_LDS 1
#else
#define CTM_ASYNC_LDS 0
#endif

static __device__ __forceinline__ void stage_b128(const float* g, float* l) {
#if CTM_ASYNC_LDS
  // builtin signature (from hipcc diagnostic): (global int4*, lds int4*, imm
  // offset, cpol) -- non-const, vector-typed pointers.
  __builtin_amdgcn_global_load_async_to_lds_b128(
      (__attribute__((address_space(1))) v4i*)g,
      (__attribute__((address_space(3))) v4i*)l, 0, 0);
#else
  *(float4*)l = *(const float4*)g;
#endif
}

static __device__ __forceinline__ void stage_wait() {
#if CTM_ASYNC_LDS
  __builtin_amdgcn_s_wait_asynccnt(0);
#endif
  __syncthreads();
}

// ---------------------------------------------------------------------------
// K1: per-token statistics.  One block (128 thr) per token.
// inv_rms = rsqrt(mean(x^2)+eps); nov_raw = mean((x*inv - x0)^2);
// sal_raw = mean(x^2)*inv^2; gdot = (x . w_gate) * inv
// ---------------------------------------------------------------------------
__global__ void k_stats(const float* __restrict__ x, const float* __restrict__ x0,
                        const float* __restrict__ wgate,
                        float* __restrict__ inv, float* __restrict__ nov,
                        float* __restrict__ sal, float* __restrict__ gdot) {
  const int tok = blockIdx.x;
  const int tid = threadIdx.x;
  const size_t base = (size_t)tok * CD;
  float sxx = 0.f, sxx0 = 0.f, sx0x0 = 0.f, sxg = 0.f;
  for (int i = tid; i < CD; i += 128) {
    float xv = x[base + i];
    float zv = x0[base + i];
    float gv = wgate[i];
    sxx   += xv * xv;
    sxx0  += xv * zv;
    sx0x0 += zv * zv;
    sxg   += xv * gv;
  }
  __shared__ float r0[128], r1[128], r2[128], r3[128];
  r0[tid] = sxx; r1[tid] = sxx0; r2[tid] = sx0x0; r3[tid] = sxg;
  __syncthreads();
  for (int off = 64; off > 0; off >>= 1) {
    if (tid < off) {
      r0[tid] += r0[tid + off];
      r1[tid] += r1[tid + off];
      r2[tid] += r2[tid + off];
      r3[tid] += r3[tid + off];
    }
    __syncthreads();
  }
  if (tid == 0) {
    float ssq = r0[0];
    float iv  = rsqrtf(ssq / (float)CD + FEPS);
    float iv2 = iv * iv;
    inv[tok]  = iv;
    sal[tok]  = ssq * iv2 / (float)CD;
    nov[tok]  = (ssq * iv2 - 2.f * iv * r1[0] + r2[0]) / (float)CD;
    gdot[tok] = r3[0] * iv;
  }
}

// ---------------------------------------------------------------------------
// K2: running-mean normalization -> gate_signal.  One wave per batch.
// Wave32 inclusive scan (shfl_up), carried across 64 chunks of 32.
// ---------------------------------------------------------------------------
__global__ void k_gate(const float* __restrict__ nov, const float* __restrict__ sal,
                       float* __restrict__ gs) {
  const int b = blockIdx.x;
  const int lane = threadIdx.x;  // 0..31
  float cn = 0.f, cs = 0.f;
  for (int chunk = 0; chunk < CT / 32; ++chunk) {
    const int t = chunk * 32 + lane;
    float vn = nov[b * CT + t];
    float vs = sal[b * CT + t];
    float sn = vn, ss = vs;
    #pragma unroll
    for (int off = 1; off < 32; off <<= 1) {
      float tn = __shfl_up(sn, off, 32);
      float ts = __shfl_up(ss, off, 32);
      if (lane >= off) { sn += tn; ss += ts; }
    }
    float cumn = cn + sn;
    float cums = cs + ss;
    float steps = (float)(t + 1);
    float g = vn / fmaxf(cumn / steps, 1e-6f) + vs / fmaxf(cums / steps, 1e-6f);
    gs[b * CT + t] = g;
    cn += __shfl(sn, 31, 32);
    cs += __shfl(ss, 31, 32);
  }
}

// ---------------------------------------------------------------------------
// K3/K5: Y[M,N] = (rowscale .* X)[M,K] @ W[N,K]^T via WMMA f32 16x16x4.
// 4-wave workgroup: one 16-row M tile staged to LDS (async b128, K chunks of
// 256, row pad of 4 floats -> ds_load_b64 bank-conflict free), each wave owns
// one 16-wide N tile (col tile = blockIdx.y*4 + wave).
// A 16x4 frag: lanes 0-15 = rows, halves hold K {0,1}/{2,3}.
// B 4x16 frag: lanes hold N, halves/VGPRs hold K (W row-major [N,K]).
// ---------------------------------------------------------------------------
__global__ void k_gemm_rs(const float* __restrict__ X, const float* __restrict__ rowscale,
                          const float* __restrict__ W, float* __restrict__ Y,
                          int K, int N) {
  const int tid  = threadIdx.x;       // 0..127
  const int wave = tid >> 5;
  const int lane = tid & 31;
  const int half = lane >> 4;
  const int lrow = lane & 15;
  const int row0 = blockIdx.x * 16;
  const int col0 = (blockIdx.y * 4 + wave) * 16;
  const int n = col0 + lrow;
  const float scale = rowscale[row0 + lrow];

  __shared__ float At[16][KC + KPAD];

  v8f acc = {};
  for (int kc = 0; kc < K; kc += KC) {
    // stage 16 x KC A tile: 1024 b128 quads, 8 per thread
    #pragma unroll
    for (int q = tid; q < 16 * (KC / 4); q += 128) {
      const int r  = q >> 6;             // KC/4 = 64 quads per row
      const int kq = (q & 63) << 2;
      stage_b128(X + (size_t)(row0 + r) * K + kc + kq, &At[r][kq]);
    }
    stage_wait();

    const float* bp = W + (size_t)n * K + kc + 2 * half;
    #pragma unroll 8
    for (int k = 0; k < KC; k += 4) {
      __builtin_prefetch(bp + k + 256, 0, 1);
      v2f a = *(const v2f*)&At[lrow][k + 2 * half];
      a *= scale;
      v2f bv = *(const v2f*)(bp + k);
      acc = wmma_f32_16x16x4(a, bv, acc);
    }
    __syncthreads();   // before next chunk overwrites At
  }
  // C/D layout: VGPR i -> (M = row0 + i + 8*half, N = col0 + (lane&15))
  #pragma unroll
  for (int i = 0; i < 8; ++i)
    Y[(size_t)(row0 + i + 8 * half) * N + col0 + lrow] = acc[i];
}

// ---------------------------------------------------------------------------
// K4: gating + softmaxes + slot_mix fold.  One block (64 thr) per token.
// token_gate = sigmoid(gdot + gs); w = softmax(sl + eg*gs)*token_gate;
// r' = slot_mix^T @ softmax(sl)   (so context = sum_s r'[s]*state[s,:]).
// ---------------------------------------------------------------------------
__global__ void k_softmax(const float* __restrict__ sl, const float* __restrict__ gs,
                          const float* __restrict__ gdot, const float* __restrict__ eg,
                          const float* __restrict__ smix,
                          float* __restrict__ wout, float* __restrict__ rpout,
                          float* __restrict__ tgout) {
  const int tok = blockIdx.x;
  const int s = threadIdx.x;  // 0..63
  __shared__ float red[CS];
  __shared__ float rbuf[CS];

  const float g = gs[tok];
  const float tgate = 1.f / (1.f + __expf(-(gdot[tok] + g)));
  const float lg = sl[(size_t)tok * CS + s];
  const float wl = lg + eg[s] * g;

  red[s] = lg; __syncthreads();
  for (int off = 32; off > 0; off >>= 1) {
    if (s < off) red[s] = fmaxf(red[s], red[s + off]);
    __syncthreads();
  }
  const float mx1 = red[0]; __syncthreads();
  float e1 = __expf(lg - mx1);
  red[s] = e1; __syncthreads();
  for (int off = 32; off > 0; off >>= 1) {
    if (s < off) red[s] += red[s + off];
    __syncthreads();
  }
  const float rw = e1 / red[0]; __syncthreads();

  red[s] = wl; __syncthreads();
  for (int off = 32; off > 0; off >>= 1) {
    if (s < off) red[s] = fmaxf(red[s], red[s + off]);
    __syncthreads();
  }
  const float mx2 = red[0]; __syncthreads();
  float e2 = __expf(wl - mx2);
  red[s] = e2; __syncthreads();
  for (int off = 32; off > 0; off >>= 1) {
    if (s < off) red[s] += red[s + off];
    __syncthreads();
  }
  const float ww = (e2 / red[0]) * tgate;

  rbuf[s] = rw; __syncthreads();
  float rp = 0.f;
  #pragma unroll 8
  for (int i = 0; i < CS; ++i) rp += rbuf[i] * smix[i * CS + s];

  wout[(size_t)tok * CS + s]  = ww;
  rpout[(size_t)tok * CS + s] = rp;
  if (s == 0) tgout[tok] = tgate;
}

// ---------------------------------------------------------------------------
// K6: per-chunk partial sums  P[b,c][s,d] = sum_{t in chunk} w[t,s]*V[t,d]
// as WMMA GEMM with M=s, N=d, K=t.  Strided scalar loads build fragments.
// ---------------------------------------------------------------------------
__global__ void k_chunk_gemm(const float* __restrict__ w, const float* __restrict__ V,
                             float* __restrict__ Pn) {
  const int chunk = blockIdx.x;           // b*NC + c
  const int b = chunk / NC;
  const int c = chunk % NC;
  const int lane = threadIdx.x;
  const int half = lane >> 4;
  const int lrow = lane & 15;
  const int s0 = blockIdx.y * 16;
  const int d0 = blockIdx.z * 16;
  const int m = s0 + lrow;                // slot index for A
  const int n = d0 + lrow;                // workspace dim for B
  const size_t bt0 = (size_t)b * CT + (size_t)c * CH;
  v8f acc = {};
  for (int k = 0; k < CH; k += 4) {
    const size_t t0 = bt0 + k + 2 * half;
    v2f a, bv;
    a.x  = w[t0 * CS + m];
    a.y  = w[(t0 + 1) * CS + m];
    bv.x = V[t0 * CWD + n];
    bv.y = V[(t0 + 1) * CWD + n];
    acc = wmma_f32_16x16x4(a, bv, acc);
  }
  #pragma unroll
  for (int i = 0; i < 8; ++i)
    Pn[((size_t)chunk * CS + s0 + i + 8 * half) * CWD + d0 + lrow] = acc[i];
}

__global__ void k_chunk_den(const float* __restrict__ w, float* __restrict__ Pd) {
  const int chunk = blockIdx.x;
  const int b = chunk / NC;
  const int c = chunk % NC;
  const int s = threadIdx.x;  // 0..63
  const size_t bt0 = (size_t)b * CT + (size_t)c * CH;
  float acc = 0.f;
  for (int t = 0; t < CH; ++t) acc += w[(bt0 + t) * CS + s];
  Pd[(size_t)chunk * CS + s] = acc;
}

// ---------------------------------------------------------------------------
// K7: in-place exclusive prefix over chunks (per batch).
// ---------------------------------------------------------------------------
__global__ void k_prefix_num(float* __restrict__ Pn) {
  const int b = blockIdx.x;
  const int s = blockIdx.y;
  const int d = threadIdx.x;  // 0..255
  float run = 0.f;
  for (int c = 0; c < NC; ++c) {
    const size_t idx = (((size_t)b * NC + c) * CS + s) * CWD + d;
    float v = Pn[idx];
    Pn[idx] = run;
    run += v;
  }
}

__global__ void k_prefix_den(float* __restrict__ Pd) {
  const int b = blockIdx.x;
  const int s = threadIdx.x;  // 0..63
  float run = 0.f;
  for (int c = 0; c < NC; ++c) {
    const size_t idx = ((size_t)b * NC + c) * CS + s;
    float v = Pd[idx];
    Pd[idx] = run;
    run += v;
  }
}

// ---------------------------------------------------------------------------
// K8: in-chunk sequential sweep.  One block (256 thr, thread = WD column d)
// per (batch, chunk).  num[s][d] held in 64 VGPRs per lane; den + r'/den
// coefficients in LDS.  Exclusive (shift-right) semantics: context for t is
// read before adding token t's rank-1 update.
// ---------------------------------------------------------------------------
__global__ void k_scan(const float* __restrict__ w, const float* __restrict__ rp,
                       const float* __restrict__ V, const float* __restrict__ Pn,
                       const float* __restrict__ Pd, float* __restrict__ ctx) {
  const int chunk = blockIdx.x;  // b*NC + c
  const int b = chunk / NC;
  const int c = chunk % NC;
  const int d = threadIdx.x;     // 0..255

  float num[CS];
  #pragma unroll
  for (int s = 0; s < CS; ++s)
    num[s] = Pn[((size_t)chunk * CS + s) * CWD + d];

  __shared__ float den[CS];
  __shared__ float coef[CS];
  if (d < CS) den[d] = Pd[(size_t)chunk * CS + d];
  __syncthreads();

  for (int tt = 0; tt < CH; ++tt) {
    const size_t bt = (size_t)b * CT + (size_t)c * CH + tt;
    const float* wrow = w + bt * CS;
    if (d < CS) {
      float dd = den[d];
      coef[d] = rp[bt * CS + d] / fmaxf(dd, 1e-6f);
      den[d] = dd + wrow[d];      // only thread d touches den[d]
    }
    __syncthreads();
    float acc = 0.f;
    #pragma unroll
    for (int s = 0; s < CS; ++s) acc += coef[s] * num[s];
    ctx[bt * CWD + d] = acc;
    const float vtd = V[bt * CWD + d];
    #pragma unroll
    for (int s = 0; s < CS; ++s) num[s] += wrow[s] * vtd;
    __syncthreads();              // protect coef before next overwrite
  }
}

// ---------------------------------------------------------------------------
// K9: out = x + ctm_scale * token_gate * (ctx @ W_model^T), WMMA f32.
// 4-wave workgroup, A tile (16 x 256) staged once to LDS (async), each wave
// serves 16-wide N tiles (col tile = blockIdx.y*4 + wave); K = WD = 256.
// ---------------------------------------------------------------------------
__global__ void k_gemm_model(const float* __restrict__ ctxv, const float* __restrict__ Wm,
                             const float* __restrict__ x, const float* __restrict__ tg,
                             const float* __restrict__ cscale, float* __restrict__ out) {
  const int tid  = threadIdx.x;       // 0..127
  const int wave = tid >> 5;
  const int lane = tid & 31;
  const int half = lane >> 4;
  const int lrow = lane & 15;
  const int row0 = blockIdx.x * 16;
  const int col0 = (blockIdx.y * 4 + wave) * 16;
  const int n = col0 + lrow;

  __shared__ float At[16][KC + KPAD];   // KC == CWD == 256
  #pragma unroll
  for (int q = tid; q < 16 * (CWD / 4); q += 128) {
    const int r  = q >> 6;
    const int kq = (q & 63) << 2;
    stage_b128(ctxv + (size_t)(row0 + r) * CWD + kq, &At[r][kq]);
  }
  stage_wait();

  const float* bp = Wm + (size_t)n * CWD + 2 * half;
  v8f acc = {};
  #pragma unroll 8
  for (int k = 0; k < CWD; k += 4) {
    __builtin_prefetch(bp + k + 128, 0, 1);
    v2f a  = *(const v2f*)&At[lrow][k + 2 * half];
    v2f bv = *(const v2f*)(bp + k);
    acc = wmma_f32_16x16x4(a, bv, acc);
  }
  #pragma unroll
  for (int i = 0; i < 8; ++i) {
    const int mr = row0 + i + 8 * half;
    const int nc = col0 + lrow;
    const size_t idx = (size_t)mr * CD + nc;
    out[idx] = x[idx] + cscale[nc] * tg[mr] * acc[i];
  }
}

// ---------------------------------------------------------------------------
// Launch.  Workspace layout (floats):
//   inv[M] nov[M] sal[M] gdot[M] gs[M] tg[M]
//   sl[M*S] w[M*S] rp[M*S] V[M*WD] ctx[M*WD] Pn[B*NC*S*WD] Pd[B*NC*S]
// Total = 3,434,496 floats ~= 13.8 MB  (fits comfortably in L2: 192 MB).
// ---------------------------------------------------------------------------
extern "C" void kernel_launch(void* const* d_in, const int* in_sizes, int n_in,
                              void* d_out, int out_size, void* d_ws, size_t ws_size,
                              hipStream_t stream) {
  const float* x      = (const float*)d_in[0];
  const float* x0     = (const float*)d_in[1];
  const float* Wsel   = (const float*)d_in[2];
  const float* Wws    = (const float*)d_in[3];
  const float* Wmod   = (const float*)d_in[4];
  const float* Wgate  = (const float*)d_in[5];
  const float* smix   = (const float*)d_in[6];
  const float* egain  = (const float*)d_in[7];
  const float* cscale = (const float*)d_in[8];
  float* out = (float*)d_out;

  float* ws = (float*)d_ws;
  float* inv  = ws;                 // M
  float* nov  = inv  + CM;          // M
  float* sal  = nov  + CM;          // M
  float* gdot = sal  + CM;          // M
  float* gs   = gdot + CM;          // M
  float* tg   = gs   + CM;          // M
  float* sl   = tg   + CM;          // M*S
  float* wbuf = sl   + (size_t)CM * CS;   // M*S
  float* rpb  = wbuf + (size_t)CM * CS;   // M*S
  float* Vb   = rpb  + (size_t)CM * CS;   // M*WD
  float* ctx  = Vb   + (size_t)CM * CWD;  // M*WD
  float* Pn   = ctx  + (size_t)CM * CWD;  // B*NC*S*WD
  float* Pd   = Pn   + (size_t)CB * NC * CS * CWD;  // B*NC*S

  k_stats<<<CM, 128, 0, stream>>>(x, x0, Wgate, inv, nov, sal, gdot);
  k_gate<<<CB, 32, 0, stream>>>(nov, sal, gs);
  k_gemm_rs<<<dim3(CM / 16, CS / 64), 128, 0, stream>>>(x, inv, Wsel, sl, CD, CS);
  k_softmax<<<CM, CS, 0, stream>>>(sl, gs, gdot, egain, smix, wbuf, rpb, tg);
  k_gemm_rs<<<dim3(CM / 16, CWD / 64), 128, 0, stream>>>(x, inv, Wws, Vb, CD, CWD);
  k_chunk_gemm<<<dim3(CB * NC, CS / 16, CWD / 16), 32, 0, stream>>>(wbuf, Vb, Pn);
  k_chunk_den<<<CB * NC, CS, 0, stream>>>(wbuf, Pd);
  k_prefix_num<<<dim3(CB, CS), CWD, 0, stream>>>(Pn);
  k_prefix_den<<<CB, CS, 0, stream>>>(Pd);
  k_scan<<<CB * NC, CWD, 0, stream>>>(wbuf, rpb, Vb, Pn, Pd, ctx);
  k_gemm_model<<<dim3(CM / 16, CD / 64), 128, 0, stream>>>(ctx, Wmod, x, tg, cscale, out);
}